// ErrorInterpolate_80187039416367
// MI455X (gfx1250) — compile-verified
//
#include <hip/hip_runtime.h>
#include <hip/hip_bf16.h>

// ---------------------------------------------------------------------------
// KNN (K=32) inverse-distance-squared interpolation, gfx1250 (CDNA5, wave32).
//
// Phase 1: distances via V_WMMA_F32_16X16X4_F32.
//   A (16x4, per query row m): [hx, hy, hz, 1]
//   B (4x16, per source col n): [-2*lx, -2*ly, -2*lz, ||l||^2]
//   C (16x16): ||h||^2 broadcast per row
//   => D = ||h||^2 - 2 h.l + ||l||^2  (exact squared distance, f32)
// One wave owns 16 query rows; each row's top-32 list is one entry per lane.
// A wave32 ballot prefilters candidates against the row's current 32nd-best,
// so the expensive replace+argmax path runs only ~K*ln(N/K) times per row.
//
// pos_l is staged into LDS in 12KB tiles with GLOBAL_LOAD_ASYNC_TO_LDS_B128
// (ASYNCcnt path, no VGPR round-trip), synchronized with s_wait_asynccnt.
//
// Phase 2: per row, broadcast (w_k, idx_k) with v_readlane; 32 lanes gather
// float4 feature slices of x (L2-resident, 4 MB) and accumulate.
// ---------------------------------------------------------------------------

typedef __attribute__((ext_vector_type(2))) float v2f;
typedef __attribute__((ext_vector_type(8))) float v8f;

#define WAVES_PER_BLOCK 4
#define THREADS (WAVES_PER_BLOCK * 32)
#define ROWS_PER_WAVE 16
#define STAGE 1024          // source points staged into LDS per iteration
#define KNN 32
#define FEAT 128
#define BIGF 3.0e38f

__device__ __forceinline__ float shflf(float v, int src) {
  return __int_as_float(__builtin_amdgcn_ds_bpermute(src << 2, __float_as_int(v)));
}
__device__ __forceinline__ int shfli(int v, int src) {
  return __builtin_amdgcn_ds_bpermute(src << 2, v);
}

// Wave-wide argmax of (value, lane-id payload); result uniform in all lanes.
__device__ __forceinline__ void wave_argmax(float& mv, int& ml, int lane) {
#pragma unroll
  for (int off = 16; off >= 1; off >>= 1) {
    float ov = shflf(mv, lane ^ off);
    int   ol = shfli(ml, lane ^ off);
    if (ov > mv || (ov == mv && ol > ml)) { mv = ov; ml = ol; }
  }
}

// Replace current worst list entry with (cd, ci) if it improves the list,
// then re-establish (max value, owning lane). cd/ci are wave-uniform.
__device__ __forceinline__ void topk_insert(float& ed, int& ei, float& mx, int& mxl,
                                            float cd, int ci, int lane) {
  if (cd < mx) {
    if (lane == mxl) { ed = cd; ei = ci; }
    float mv = ed; int ml = lane;
    wave_argmax(mv, ml, lane);
    mx = mv; mxl = ml;
  }
}

// Process one 16-source chunk: one WMMA -> 16x16 distance tile -> ballot-
// prefiltered top-K merge. FULL=true removes all tail guards (hot path).
template <bool FULL>
__device__ __forceinline__ void process_chunk(
    const float* __restrict__ s_pos, int c, int nsrc, int colbase, int lane,
    v2f A, v8f Cacc, float* td, int* ti, float* tmax, int* tml) {
  unsigned vm16 = 0xFFFFu;
  int sidx = c + (lane & 15);
  if (!FULL) {
    const int rem = nsrc - c;
    vm16 = (1u << rem) - 1u;
    if (sidx >= nsrc) sidx = nsrc - 1;
  }
  const float qx = s_pos[sidx * 3 + 0];
  const float qy = s_pos[sidx * 3 + 1];
  const float qz = s_pos[sidx * 3 + 2];
  const float l2 = qx * qx + qy * qy + qz * qz;
  v2f B;
  B.x = (lane < 16) ? (-2.0f * qx) : (-2.0f * qz); // K=0 / K=2
  B.y = (lane < 16) ? (-2.0f * qy) : l2;           // K=1 / K=3

  // 16x16 tile of squared distances in one matrix op.
  v8f Dm = __builtin_amdgcn_wmma_f32_16x16x4_f32(
      false, A, false, B, (short)0, Cacc, false, false);

#pragma unroll
  for (int r = 0; r < 8; ++r) {
    // Row r lives in lanes 0-15 of Dm[r]; row r+8 in lanes 16-31.
    const float rm = (lane < 16) ? tmax[r] : tmax[r + 8];
    const unsigned mask = __builtin_amdgcn_ballot_w32(Dm[r] < rm);
    unsigned lo = (mask & 0xFFFFu) & vm16;
    while (lo) {
      const int s = __builtin_ctz(lo); lo &= lo - 1;
      topk_insert(td[r], ti[r], tmax[r], tml[r],
                  shflf(Dm[r], s), colbase + s, lane);
    }
    unsigned hi = (mask >> 16) & vm16;
    while (hi) {
      const int s = __builtin_ctz(hi); hi &= hi - 1;
      topk_insert(td[r + 8], ti[r + 8], tmax[r + 8], tml[r + 8],
                  shflf(Dm[r], 16 + s), colbase + s, lane);
    }
  }
}

__global__ __launch_bounds__(THREADS)
void knn_interp_kernel(const float* __restrict__ x,
                       const float* __restrict__ pos_l,
                       const float* __restrict__ pos_h,
                       float* __restrict__ y,
                       int N_l, int N_h) {
  // NOTE: s_pos is the only LDS object in this kernel, so its LDS byte
  // offset is 0; the async-copy DS addresses below are formed directly.
  __shared__ float s_pos[STAGE * 3];

  const int lane = threadIdx.x & 31;
  const int wave = threadIdx.x >> 5;
  const int tid  = threadIdx.x;
  const int row_base = (blockIdx.x * WAVES_PER_BLOCK + wave) * ROWS_PER_WAVE;

  // ---- A matrix (16x4 f32). ISA layout: lanes 0-15 hold K=0,1 in the two
  // VGPRs; lanes 16-31 hold K=2,3. Row M = lane&15.
  int mrow = row_base + (lane & 15);
  if (mrow >= N_h) mrow = N_h - 1;          // clamp (keeps EXEC all-ones)
  const float hx = pos_h[(size_t)mrow * 3 + 0];
  const float hy = pos_h[(size_t)mrow * 3 + 1];
  const float hz = pos_h[(size_t)mrow * 3 + 2];
  const float h2 = hx * hx + hy * hy + hz * hz;

  v2f A;
  A.x = (lane < 16) ? hx : hz;              // K=0 / K=2
  A.y = (lane < 16) ? hy : 1.0f;            // K=1 / K=3

  // ---- C accumulator: C[r] = ||h||^2 of row (r + 8*(lane>=16)); loop-invariant.
  v8f Cacc;
#pragma unroll
  for (int r = 0; r < 8; ++r)
    Cacc[r] = shflf(h2, r + ((lane >> 4) << 3));

  // ---- per-row top-K state: one list entry per lane; max tracked uniformly.
  float td[ROWS_PER_WAVE]; int ti[ROWS_PER_WAVE];
  float tmax[ROWS_PER_WAVE]; int tml[ROWS_PER_WAVE];
#pragma unroll
  for (int m = 0; m < ROWS_PER_WAVE; ++m) {
    td[m] = BIGF; ti[m] = 0; tmax[m] = BIGF; tml[m] = 31;
  }

  for (int sc = 0; sc < N_l; sc += STAGE) {
    __syncthreads();                        // LDS free of readers
    const int nsrc = min(STAGE, N_l - sc);

    if (nsrc == STAGE) {
      // Full 12KB stage: async copy, 128-bit granularity, 6 per thread.
      for (int j = tid; j < (STAGE * 3) / 4; j += THREADS) {
        const float* gp = pos_l + (size_t)sc * 3 + (size_t)j * 4;
        const unsigned ldsb = (unsigned)(j * 16);
        asm volatile("global_load_async_to_lds_b128 %0, %1, off"
                     :: "v"(ldsb), "v"(gp) : "memory");
      }
    } else {
      // Tail stage: per-dword async copy with exact bound.
      for (int i = tid; i < nsrc * 3; i += THREADS) {
        const float* gp = pos_l + (size_t)sc * 3 + i;
        const unsigned ldsb = (unsigned)(i * 4);
        asm volatile("global_load_async_to_lds_b32 %0, %1, off"
                     :: "v"(ldsb), "v"(gp) : "memory");
      }
    }
    if (sc + STAGE < N_l)   // cover next 12KB stage with global_prefetch
      __builtin_prefetch(pos_l + (size_t)(sc + STAGE) * 3 + tid * 24, 0, 1);

    asm volatile("s_wait_asynccnt 0" ::: "memory");
    __syncthreads();                        // stage visible to all waves

    const int nfull = nsrc & ~15;
    const int colbase0 = sc;
    for (int c = 0; c < nfull; c += 16)
      process_chunk<true>(s_pos, c, nsrc, colbase0 + c, lane, A, Cacc,
                          td, ti, tmax, tml);
    if (nfull < nsrc)
      process_chunk<false>(s_pos, nfull, nsrc, colbase0 + nfull, lane, A, Cacc,
                           td, ti, tmax, tml);
  }

  // ---- Phase 2: weighted feature gather. Lane k owns list entry k.
  const float4* xv = (const float4*)x;      // x row = 32 float4
#pragma unroll
  for (int m = 0; m < ROWS_PER_WAVE; ++m) {
    const float w = 1.0f / fmaxf(td[m], 1e-16f);
    float sw = w;
#pragma unroll
    for (int off = 16; off >= 1; off >>= 1) sw += shflf(sw, lane ^ off);

    float ax = 0.f, ay = 0.f, az = 0.f, aw = 0.f;
    for (int k = 0; k < KNN; ++k) {
      const float wk = __int_as_float(
          __builtin_amdgcn_readlane(__float_as_int(w), k));
      const int ik = __builtin_amdgcn_readlane(ti[m], k);
      const float4 xr = xv[(size_t)ik * (FEAT / 4) + lane];
      ax += wk * xr.x; ay += wk * xr.y; az += wk * xr.z; aw += wk * xr.w;
    }

    const int row = row_base + m;
    if (row < N_h) {
      const float inv = 1.0f / sw;
      float4 o; o.x = ax * inv; o.y = ay * inv; o.z = az * inv; o.w = aw * inv;
      ((float4*)(y + (size_t)row * FEAT))[lane] = o;
    }
  }
}

extern "C" void kernel_launch(void* const* d_in, const int* in_sizes, int n_in,
                              void* d_out, int out_size, void* d_ws, size_t ws_size,
                              hipStream_t stream) {
  const float* x     = (const float*)d_in[0];   // [N_l, 128]
  const float* pos_l = (const float*)d_in[1];   // [N_l, 3]
  const float* pos_h = (const float*)d_in[2];   // [N_h, 3]
  float* y = (float*)d_out;                     // [N_h, 128]

  const int N_l = in_sizes[1] / 3;
  const int N_h = in_sizes[2] / 3;

  const int rows_per_block = WAVES_PER_BLOCK * ROWS_PER_WAVE; // 64
  const int blocks = (N_h + rows_per_block - 1) / rows_per_block;
  knn_interp_kernel<<<blocks, THREADS, 0, stream>>>(x, pos_l, pos_h, y, N_l, N_h);
}